// CliffordSpectralConv3d_49409303773607
// MI455X (gfx1250) — compile-verified
//
#include <hip/hip_runtime.h>

// CliffordSpectralConv3d for MI455X (gfx1250, wave32, WMMA + TDM).
// Pipeline:
//   1) pruned DFT along W  (48 -> 16 modes)  : WMMA f32 16x16x4 GEMM
//   2) pruned DFT along H  (48 -> 16)        : WMMA, LDS staged via TDM (1D, 6KB)
//   3) pruned DFT along D  (48 -> 16)        : WMMA, writes mv layout [mode][b*192+i]
//   4) per-mode Clifford 192x192 matmul      : VALU (1.2 GFLOP, both operands
//                                              mode-dependent -> no WMMA reuse)
//   5) inverse DFT along D (16 -> 48)        : WMMA, LDS staged via TDM (2D x2)
//   6) inverse DFT along H (16 -> 48)        : WMMA, LDS staged via TDM (1D, 2KB)
//   7) inverse DFT along W (16 -> 48) + scatter to (B,D,H,W,Cout,8) output
// All DFT stages: complex GEMM = 4 real V_WMMA_F32_16X16X4_F32 per K-chunk.

#define S   48
#define NM  16
#define NB  4
#define CIN 24
#define COUT 24

typedef __attribute__((ext_vector_type(2))) float v2f;
typedef __attribute__((ext_vector_type(8))) float v8f;
typedef unsigned int u32;
typedef __attribute__((ext_vector_type(4))) unsigned int v4u;
typedef __attribute__((ext_vector_type(4))) int v4i;
typedef __attribute__((ext_vector_type(8))) int v8i;

static __device__ __forceinline__ v8f wmma4(v2f a, v2f b, v8f c) {
  // D = A(16x4 f32) * B(4x16 f32) + C(16x16 f32)
  return __builtin_amdgcn_wmma_f32_16x16x4_f32(false, a, false, b, (short)0, c,
                                               false, false);
}

// ----------------------------------------------------------- TDM helpers ----
#if __has_builtin(__builtin_amdgcn_tensor_load_to_lds)
#define HAS_TDM 1
#if __has_include(<hip/amd_detail/amd_gfx1250_TDM.h>)
#define TDM_6ARG 1
#endif
#endif

#ifdef HAS_TDM
static __device__ __forceinline__ void tdm_issue(v4u g0, v8i g1) {
  v4i z4 = {0, 0, 0, 0};
#ifdef TDM_6ARG
  v8i z8 = {0, 0, 0, 0, 0, 0, 0, 0};
  __builtin_amdgcn_tensor_load_to_lds(g0, g1, z4, z4, z8, 0);
#else
  __builtin_amdgcn_tensor_load_to_lds(g0, g1, z4, z4, 0);
#endif
}

// 1-D contiguous copy: n8 elements of 8 bytes, global -> LDS.
static __device__ __forceinline__ void tdm_load_1d(u32 lds_byte,
                                                   unsigned long long gaddr,
                                                   u32 n8) {
  v4u g0;
  g0[0] = 1u;                                     // count=1, user descriptor
  g0[1] = lds_byte;                               // lds_addr
  g0[2] = (u32)(gaddr & 0xffffffffull);           // global_addr[31:0]
  g0[3] = (u32)((gaddr >> 32) & 0x01ffffffull) | (2u << 30);  // [56:32]|type=2
  v8i g1;
  g1[0] = (int)(3u << 16);                        // data_size = 8B
  g1[1] = (int)((n8 & 0xffffu) << 16);            // tensor_dim0[15:0]
  g1[2] = (int)((n8 >> 16) | (1u << 16));         // tensor_dim0 hi | tensor_dim1=1
  g1[3] = (int)((n8 & 0xffffu) << 16);            // tile_dim0
  g1[4] = 0;                                      // tile_dim1=0 (1-D tile)
  g1[5] = (int)n8;                                // tensor_dim0_stride
  g1[6] = 0;
  g1[7] = 0;
  tdm_issue(g0, g1);
}

// 2-D copy: rows x cols elements of 4 bytes, row stride (elements) in global.
static __device__ __forceinline__ void tdm_load_2d(u32 lds_byte,
                                                   unsigned long long gaddr,
                                                   u32 cols, u32 rows,
                                                   u32 stride) {
  v4u g0;
  g0[0] = 1u;
  g0[1] = lds_byte;
  g0[2] = (u32)(gaddr & 0xffffffffull);
  g0[3] = (u32)((gaddr >> 32) & 0x01ffffffull) | (2u << 30);
  v8i g1;
  g1[0] = (int)(2u << 16);                        // data_size = 4B
  g1[1] = (int)((cols & 0xffffu) << 16);          // tensor_dim0[15:0]
  g1[2] = (int)((cols >> 16) | ((rows & 0xffffu) << 16));  // dim0 hi | dim1 lo
  g1[3] = (int)((rows >> 16) | ((cols & 0xffffu) << 16));  // dim1 hi | tile_dim0
  g1[4] = (int)(rows & 0xffffu);                  // tile_dim1 | tile_dim2=0
  g1[5] = (int)stride;                            // tensor_dim0_stride lo
  g1[6] = 0;
  g1[7] = 0;
  tdm_issue(g0, g1);
}
#endif  // HAS_TDM

// ---------------------------------------------------------------- twiddles --
__global__ void k_twiddle(float* __restrict__ tfre, float* __restrict__ tfim,
                          float* __restrict__ tire, float* __restrict__ tiim) {
  const float twopi = 6.283185307179586f;
  // forward: Tf[k][n] = exp(-2*pi*i * i1[k]*n / 48), k in 16 kept modes
  for (int t = threadIdx.x; t < 16 * 48; t += blockDim.x) {
    int k = t / 48, n = t % 48;
    int kk = (k < 8) ? k : (k + 32);               // {0..7, 40..47}
    float th = twopi * (float)((kk * n) % 48) / 48.0f;
    tfre[t] = cosf(th);
    tfim[t] = -sinf(th);
  }
  // inverse: Ti[m][k] = exp(+2*pi*i * m*i1[k] / 48) / 48
  for (int t = threadIdx.x; t < 48 * 16; t += blockDim.x) {
    int m = t / 16, k = t % 16;
    int kk = (k < 8) ? k : (k + 32);
    float th = twopi * (float)((m * kk) % 48) / 48.0f;
    tire[t] = cosf(th) * (1.0f / 48.0f);
    tiim[t] = sinf(th) * (1.0f / 48.0f);
  }
}

// ------------------------------------------------------- stage 1: DFT on W --
// x: (B,48,48,48,Cin,8) f32.  field f: re=ch f, im=ch 7-f.
// out bufA[((f*4+b)*24+c)*48+d][h][kw]  (complex)
__global__ void __launch_bounds__(128)
k_fwd_w(const float* __restrict__ x, const float* __restrict__ tfre,
        const float* __restrict__ tfim, float2* __restrict__ bufA) {
  __shared__ float tw_re[16][48], tw_im[16][48];
  __shared__ float dre[4][48][16], dim_[4][48][16];
  for (int t = threadIdx.x; t < 768; t += blockDim.x) {
    tw_re[t / 48][t % 48] = tfre[t];
    tw_im[t / 48][t % 48] = tfim[t];
  }
  const int wv = threadIdx.x >> 5;
  const int lane = threadIdx.x & 31;
  long wid = (long)blockIdx.x * 4 + wv;       // (((f*4+b)*24+c)*48+d)*3+hg
  int hg = (int)(wid % 3); long r0 = wid / 3;
  int d = (int)(r0 % 48); r0 /= 48;
  int c = (int)(r0 % 24); r0 /= 24;
  int b = (int)(r0 % 4);
  int f = (int)(r0 / 4);
  for (int t = lane; t < 768; t += 32) {       // 48 w  x 16 h-lines (scattered)
    int wk = t % 48, col = t / 48;
    int h = hg * 16 + col;
    long xi = (((((long)b * S + d) * S + h) * S + wk) * CIN + c) * 8;
    dre[wv][wk][col] = x[xi + f];
    dim_[wv][wk][col] = x[xi + (7 - f)];
  }
  __syncthreads();
  const int rowA = lane & 15;
  const int kA = (lane >> 4) << 1;
  const int colB = lane & 15;
  v8f cre = {}; v8f cim = {};
#pragma unroll
  for (int k0 = 0; k0 < 48; k0 += 4) {
    v2f are = { tw_re[rowA][k0 + kA], tw_re[rowA][k0 + kA + 1] };
    v2f aim = { tw_im[rowA][k0 + kA], tw_im[rowA][k0 + kA + 1] };
    v2f ain = -aim;
    v2f bre = { dre[wv][k0 + kA][colB], dre[wv][k0 + kA + 1][colB] };
    v2f bim = { dim_[wv][k0 + kA][colB], dim_[wv][k0 + kA + 1][colB] };
    cre = wmma4(are, bre, cre);
    cre = wmma4(ain, bim, cre);
    cim = wmma4(are, bim, cim);
    cim = wmma4(aim, bre, cim);
  }
  long fbc = ((long)f * 4 + b) * 24 + c;
  const int hi = (lane >> 4) << 3;
  const int N = lane & 15;
#pragma unroll
  for (int rr = 0; rr < 8; ++rr) {
    int M = rr + hi;                 // kw mode
    int h = hg * 16 + N;
    long oi = ((fbc * S + d) * S + h) * 16 + M;
    float2 v; v.x = cre[rr]; v.y = cim[rr];
    bufA[oi] = v;
  }
}

// ------------------------------------------------------- stage 2: DFT on H --
// bufA[line=fbc*48+d][h][kw] -> bufB[line][kh][kw]
// Per-wave input tile is 6KB contiguous: stage with the Tensor Data Mover.
__global__ void __launch_bounds__(128)
k_fwd_h(const float2* __restrict__ bufA, const float* __restrict__ tfre,
        const float* __restrict__ tfim, float2* __restrict__ bufB) {
  __shared__ float tw_re[16][48], tw_im[16][48];
  __shared__ float2 dd[4][48][16];
  const int wv = threadIdx.x >> 5;
  const int lane = threadIdx.x & 31;
  long wid = (long)blockIdx.x * 4 + wv;        // fbc*48 + d  (18432)
#ifdef HAS_TDM
  tdm_load_1d((u32)(unsigned long long)(uintptr_t)&dd[wv][0][0],
              (unsigned long long)(uintptr_t)(bufA + wid * 768), 768u);
#else
  for (int t = lane; t < 768; t += 32)
    dd[wv][t >> 4][t & 15] = bufA[wid * 768 + t];
#endif
  for (int t = threadIdx.x; t < 768; t += blockDim.x) {
    tw_re[t / 48][t % 48] = tfre[t];
    tw_im[t / 48][t % 48] = tfim[t];
  }
#ifdef HAS_TDM
  __builtin_amdgcn_s_wait_tensorcnt(0);
#endif
  __syncthreads();
  const int rowA = lane & 15;
  const int kA = (lane >> 4) << 1;
  const int colB = lane & 15;
  v8f cre = {}; v8f cim = {};
#pragma unroll
  for (int k0 = 0; k0 < 48; k0 += 4) {
    v2f are = { tw_re[rowA][k0 + kA], tw_re[rowA][k0 + kA + 1] };
    v2f aim = { tw_im[rowA][k0 + kA], tw_im[rowA][k0 + kA + 1] };
    v2f ain = -aim;
    v2f bre = { dd[wv][k0 + kA][colB].x, dd[wv][k0 + kA + 1][colB].x };
    v2f bim = { dd[wv][k0 + kA][colB].y, dd[wv][k0 + kA + 1][colB].y };
    cre = wmma4(are, bre, cre);
    cre = wmma4(ain, bim, cre);
    cim = wmma4(are, bim, cim);
    cim = wmma4(aim, bre, cim);
  }
  const int hi = (lane >> 4) << 3;
  const int N = lane & 15;
#pragma unroll
  for (int rr = 0; rr < 8; ++rr) {
    int M = rr + hi;                  // kh
    float2 v; v.x = cre[rr]; v.y = cim[rr];
    bufB[wid * 256 + M * 16 + N] = v;
  }
}

// ------------------------------------------------------- stage 3: DFT on D --
// bufB[fbc*48+d][kh][kw] -> bufC[mode][b*192+i]  (mv real layout)
__global__ void __launch_bounds__(128)
k_fwd_d(const float2* __restrict__ bufB, const float* __restrict__ tfre,
        const float* __restrict__ tfim, float* __restrict__ bufC) {
  __shared__ float tw_re[16][48], tw_im[16][48];
  __shared__ float dre[4][48][16], dim_[4][48][16];
  for (int t = threadIdx.x; t < 768; t += blockDim.x) {
    tw_re[t / 48][t % 48] = tfre[t];
    tw_im[t / 48][t % 48] = tfim[t];
  }
  const int wv = threadIdx.x >> 5;
  const int lane = threadIdx.x & 31;
  long wid = (long)blockIdx.x * 4 + wv;     // fbc*16 + kh  (6144)
  int kh = (int)(wid % 16);
  long fbc = wid / 16;
  int c = (int)(fbc % 24); long t2 = fbc / 24;
  int b = (int)(t2 % 4);
  int f = (int)(t2 / 4);
  for (int t = lane; t < 768; t += 32) {    // 48 d x 16 kw
    int dd = t >> 4, kw = t & 15;
    float2 v = bufB[fbc * 12288 + (long)dd * 256 + kh * 16 + kw];
    dre[wv][dd][kw] = v.x;
    dim_[wv][dd][kw] = v.y;
  }
  __syncthreads();
  const int rowA = lane & 15;
  const int kA = (lane >> 4) << 1;
  const int colB = lane & 15;
  v8f cre = {}; v8f cim = {};
#pragma unroll
  for (int k0 = 0; k0 < 48; k0 += 4) {
    v2f are = { tw_re[rowA][k0 + kA], tw_re[rowA][k0 + kA + 1] };
    v2f aim = { tw_im[rowA][k0 + kA], tw_im[rowA][k0 + kA + 1] };
    v2f ain = -aim;
    v2f bre = { dre[wv][k0 + kA][colB], dre[wv][k0 + kA + 1][colB] };
    v2f bim = { dim_[wv][k0 + kA][colB], dim_[wv][k0 + kA + 1][colB] };
    cre = wmma4(are, bre, cre);
    cre = wmma4(ain, bim, cre);
    cim = wmma4(are, bim, cim);
    cim = wmma4(aim, bre, cim);
  }
  const int i_re = f * 24 + c;
  const int i_im = (7 - f) * 24 + c;
  const int hi = (lane >> 4) << 3;
  const int N = lane & 15;
#pragma unroll
  for (int rr = 0; rr < 8; ++rr) {
    int kd = rr + hi;
    long mode = (long)kd * 256 + kh * 16 + N;
    bufC[mode * 768 + b * 192 + i_re] = cre[rr];
    bufC[mode * 768 + b * 192 + i_im] = cim[rr];
  }
}

// --------------------------------------------- stage 4: per-mode Clifford ---
__device__ const int g_widx[8][8] = {
  {0,1,2,3,4,5,6,7},
  {1,0,4,5,2,3,7,6},
  {2,4,0,6,1,7,3,5},
  {3,5,6,0,7,1,2,4},
  {4,2,1,7,0,6,5,3},
  {5,3,7,1,6,0,4,2},
  {6,7,3,2,5,4,0,1},
  {7,6,5,4,3,2,1,0}};
__device__ const float g_sgn[8][8] = {
  { 1, 1, 1, 1,-1,-1,-1,-1},
  { 1, 1,-1,-1, 1, 1,-1,-1},
  { 1, 1, 1,-1,-1, 1, 1, 1},
  { 1, 1, 1, 1,-1,-1,-1,-1},
  { 1, 1,-1, 1, 1,-1, 1, 1},
  { 1, 1,-1,-1, 1, 1,-1,-1},
  { 1, 1, 1,-1,-1, 1, 1, 1},
  { 1, 1,-1, 1, 1,-1, 1, 1}};

// bufC[mode][b*192+i], weights (8,24,24,4096) -> bufD[(b*192+o)*4096+mode]
__global__ void __launch_bounds__(256)
k_cliff(const float* __restrict__ bufC, const float* __restrict__ w,
        float* __restrict__ bufD) {
  __shared__ float in_s[768];
  __shared__ float w_s[4608];          // 8*24*24 weights at this mode
  const int m = blockIdx.x;
  for (int t = threadIdx.x; t < 768; t += 256)
    in_s[t] = bufC[(long)m * 768 + t];
  for (int t = threadIdx.x; t < 4608; t += 256)
    w_s[t] = w[(long)t * 4096 + m];
  __syncthreads();
  for (int u = threadIdx.x; u < 768; u += 256) {
    int b = u / 192, o = u % 192;
    int ob = o / 24, oc = o % 24;
    float acc = 0.0f;
#pragma unroll
    for (int ib = 0; ib < 8; ++ib) {
      const float* wp = &w_s[(g_widx[ob][ib] * 24 + oc) * 24];
      const float* ip = &in_s[b * 192 + ib * 24];
      float blk = 0.0f;
#pragma unroll
      for (int ic = 0; ic < 24; ++ic) blk = fmaf(wp[ic], ip[ic], blk);
      acc = fmaf(g_sgn[ob][ib], blk, acc);
    }
    bufD[((long)b * 192 + o) * 4096 + m] = acc;
  }
}

// -------------------------------------------------- stage 5: inverse on D ---
// bufD (mv) -> bufB[fbc*48+d][kh][kw].  Re/Im planes staged via 2-D TDM.
__global__ void __launch_bounds__(128)
k_inv_d(const float* __restrict__ bufD, const float* __restrict__ tire,
        const float* __restrict__ tiim, float2* __restrict__ bufB) {
  __shared__ float ti_re[48][16], ti_im[48][16];
  __shared__ float dre[4][16][16], dim_[4][16][16];
  const int wv = threadIdx.x >> 5;
  const int lane = threadIdx.x & 31;
  long wid = (long)blockIdx.x * 4 + wv;     // (((f*4+b)*24+c)*16+kh)*3+g
  int g = (int)(wid % 3); long r0 = wid / 3;
  int kh = (int)(r0 % 16); r0 /= 16;
  int c = (int)(r0 % 24); r0 /= 24;
  int b = (int)(r0 % 4);
  int f = (int)(r0 / 4);
  const int i_re = f * 24 + c;
  const int i_im = (7 - f) * 24 + c;
#ifdef HAS_TDM
  // 16 rows (kd) x 16 floats, global row stride 256 floats
  tdm_load_2d((u32)(unsigned long long)(uintptr_t)&dre[wv][0][0],
              (unsigned long long)(uintptr_t)
                  (bufD + ((long)b * 192 + i_re) * 4096 + kh * 16),
              16u, 16u, 256u);
  tdm_load_2d((u32)(unsigned long long)(uintptr_t)&dim_[wv][0][0],
              (unsigned long long)(uintptr_t)
                  (bufD + ((long)b * 192 + i_im) * 4096 + kh * 16),
              16u, 16u, 256u);
#else
  for (int t = lane; t < 256; t += 32) {    // 16 kd x 16 kw
    int kd = t >> 4, kw = t & 15;
    long mo = (long)kd * 256 + kh * 16 + kw;
    dre[wv][kd][kw] = bufD[((long)b * 192 + i_re) * 4096 + mo];
    dim_[wv][kd][kw] = bufD[((long)b * 192 + i_im) * 4096 + mo];
  }
#endif
  for (int t = threadIdx.x; t < 768; t += blockDim.x) {
    ti_re[t / 16][t % 16] = tire[t];
    ti_im[t / 16][t % 16] = tiim[t];
  }
#ifdef HAS_TDM
  __builtin_amdgcn_s_wait_tensorcnt(0);
#endif
  __syncthreads();
  const int rowA = lane & 15;
  const int kA = (lane >> 4) << 1;
  const int colB = lane & 15;
  v8f cre = {}; v8f cim = {};
#pragma unroll
  for (int k0 = 0; k0 < 16; k0 += 4) {
    v2f are = { ti_re[g * 16 + rowA][k0 + kA], ti_re[g * 16 + rowA][k0 + kA + 1] };
    v2f aim = { ti_im[g * 16 + rowA][k0 + kA], ti_im[g * 16 + rowA][k0 + kA + 1] };
    v2f ain = -aim;
    v2f bre = { dre[wv][k0 + kA][colB], dre[wv][k0 + kA + 1][colB] };
    v2f bim = { dim_[wv][k0 + kA][colB], dim_[wv][k0 + kA + 1][colB] };
    cre = wmma4(are, bre, cre);
    cre = wmma4(ain, bim, cre);
    cim = wmma4(are, bim, cim);
    cim = wmma4(aim, bre, cim);
  }
  long fbc = ((long)f * 4 + b) * 24 + c;
  const int hi = (lane >> 4) << 3;
  const int N = lane & 15;
#pragma unroll
  for (int rr = 0; rr < 8; ++rr) {
    int d = g * 16 + rr + hi;
    float2 v; v.x = cre[rr]; v.y = cim[rr];
    bufB[(fbc * S + d) * 256 + kh * 16 + N] = v;
  }
}

// -------------------------------------------------- stage 6: inverse on H ---
// bufB[line][kh][kw] -> bufA[line][h][kw].  2KB contiguous tile via 1-D TDM.
__global__ void __launch_bounds__(128)
k_inv_h(const float2* __restrict__ bufB, const float* __restrict__ tire,
        const float* __restrict__ tiim, float2* __restrict__ bufA) {
  __shared__ float ti_re[48][16], ti_im[48][16];
  __shared__ float2 dd[4][16][16];
  const int wv = threadIdx.x >> 5;
  const int lane = threadIdx.x & 31;
  long wid = (long)blockIdx.x * 4 + wv;     // (fbc*48+d)*3+hg
  int hg = (int)(wid % 3);
  long line = wid / 3;                      // fbc*48+d
#ifdef HAS_TDM
  tdm_load_1d((u32)(unsigned long long)(uintptr_t)&dd[wv][0][0],
              (unsigned long long)(uintptr_t)(bufB + line * 256), 256u);
#else
  for (int t = lane; t < 256; t += 32)
    dd[wv][t >> 4][t & 15] = bufB[line * 256 + t];
#endif
  for (int t = threadIdx.x; t < 768; t += blockDim.x) {
    ti_re[t / 16][t % 16] = tire[t];
    ti_im[t / 16][t % 16] = tiim[t];
  }
#ifdef HAS_TDM
  __builtin_amdgcn_s_wait_tensorcnt(0);
#endif
  __syncthreads();
  const int rowA = lane & 15;
  const int kA = (lane >> 4) << 1;
  const int colB = lane & 15;
  v8f cre = {}; v8f cim = {};
#pragma unroll
  for (int k0 = 0; k0 < 16; k0 += 4) {
    v2f are = { ti_re[hg * 16 + rowA][k0 + kA], ti_re[hg * 16 + rowA][k0 + kA + 1] };
    v2f aim = { ti_im[hg * 16 + rowA][k0 + kA], ti_im[hg * 16 + rowA][k0 + kA + 1] };
    v2f ain = -aim;
    v2f bre = { dd[wv][k0 + kA][colB].x, dd[wv][k0 + kA + 1][colB].x };
    v2f bim = { dd[wv][k0 + kA][colB].y, dd[wv][k0 + kA + 1][colB].y };
    cre = wmma4(are, bre, cre);
    cre = wmma4(ain, bim, cre);
    cim = wmma4(are, bim, cim);
    cim = wmma4(aim, bre, cim);
  }
  const int hi = (lane >> 4) << 3;
  const int N = lane & 15;
#pragma unroll
  for (int rr = 0; rr < 8; ++rr) {
    int h = hg * 16 + rr + hi;
    float2 v; v.x = cre[rr]; v.y = cim[rr];
    bufA[(line * S + h) * 16 + N] = v;
  }
}

// ------------------------------------- stage 7: inverse on W + scatter out --
// bufA[line][h][kw] -> out (B,48,48,48,Cout,8): re->ch f, im->ch 7-f
__global__ void __launch_bounds__(128)
k_inv_w(const float2* __restrict__ bufA, const float* __restrict__ tire,
        const float* __restrict__ tiim, float* __restrict__ out) {
  __shared__ float ti_re[48][16], ti_im[48][16];
  __shared__ float dre[4][16][16], dim_[4][16][16];
  for (int t = threadIdx.x; t < 768; t += blockDim.x) {
    ti_re[t / 16][t % 16] = tire[t];
    ti_im[t / 16][t % 16] = tiim[t];
  }
  const int wv = threadIdx.x >> 5;
  const int lane = threadIdx.x & 31;
  long wid = (long)blockIdx.x * 4 + wv;     // ((fbc*48+d)*3+hg)*3+wg
  int wg = (int)(wid % 3); long r0 = wid / 3;
  int hg = (int)(r0 % 3);
  long line = r0 / 3;                       // fbc*48+d
  int d = (int)(line % 48);
  long fbc = line / 48;
  int c = (int)(fbc % 24); long t2 = fbc / 24;
  int b = (int)(t2 % 4);
  int f = (int)(t2 / 4);
  for (int t = lane; t < 256; t += 32) {    // [kw][hcol], transposed stage
    int hcol = t >> 4, kw = t & 15;
    float2 v = bufA[(line * S + hg * 16 + hcol) * 16 + kw];
    dre[wv][kw][hcol] = v.x;
    dim_[wv][kw][hcol] = v.y;
  }
  __syncthreads();
  const int rowA = lane & 15;
  const int kA = (lane >> 4) << 1;
  const int colB = lane & 15;
  v8f cre = {}; v8f cim = {};
#pragma unroll
  for (int k0 = 0; k0 < 16; k0 += 4) {
    v2f are = { ti_re[wg * 16 + rowA][k0 + kA], ti_re[wg * 16 + rowA][k0 + kA + 1] };
    v2f aim = { ti_im[wg * 16 + rowA][k0 + kA], ti_im[wg * 16 + rowA][k0 + kA + 1] };
    v2f ain = -aim;
    v2f bre = { dre[wv][k0 + kA][colB], dre[wv][k0 + kA + 1][colB] };
    v2f bim = { dim_[wv][k0 + kA][colB], dim_[wv][k0 + kA + 1][colB] };
    cre = wmma4(are, bre, cre);
    cre = wmma4(ain, bim, cre);
    cim = wmma4(are, bim, cim);
    cim = wmma4(aim, bre, cim);
  }
  const int hi = (lane >> 4) << 3;
  const int N = lane & 15;
#pragma unroll
  for (int rr = 0; rr < 8; ++rr) {
    int ww = wg * 16 + rr + hi;            // w
    int h = hg * 16 + N;
    long oi = (((((long)b * S + d) * S + h) * S + ww) * COUT + c) * 8;
    out[oi + f] = cre[rr];
    out[oi + (7 - f)] = cim[rr];
  }
}

// ------------------------------------------------------------------ launch --
extern "C" void kernel_launch(void* const* d_in, const int* in_sizes, int n_in,
                              void* d_out, int out_size, void* d_ws,
                              size_t ws_size, hipStream_t stream) {
  (void)in_sizes; (void)n_in; (void)out_size; (void)ws_size;
  const float* x = (const float*)d_in[0];
  const float* wts = (const float*)d_in[1];
  float* out = (float*)d_out;

  float* wsf = (float*)d_ws;
  float* tfre = wsf + 0;       // 16*48
  float* tfim = wsf + 768;
  float* tire = wsf + 1536;    // 48*16
  float* tiim = wsf + 2304;

  char* base = (char*)d_ws;
  const unsigned long long offA = 16384ULL;
  const unsigned long long offB = offA + 113246208ULL;   // 4*4*24*48*48*16 cplx
  const unsigned long long offC = offB + 37748736ULL;    // 4*4*24*48*16*16 cplx
  const unsigned long long offD = offC + 12582912ULL;    // 4*192*4096 f32
  float2* bufA = (float2*)(base + offA);
  float2* bufB = (float2*)(base + offB);
  float*  bufC = (float*) (base + offC);
  float*  bufD = (float*) (base + offD);

  k_twiddle<<<1, 256, 0, stream>>>(tfre, tfim, tire, tiim);
  k_fwd_w<<<13824, 128, 0, stream>>>(x, tfre, tfim, bufA);    // 55296 waves
  k_fwd_h<<<4608, 128, 0, stream>>>(bufA, tfre, tfim, bufB);  // 18432
  k_fwd_d<<<1536, 128, 0, stream>>>(bufB, tfre, tfim, bufC);  // 6144
  k_cliff<<<4096, 256, 0, stream>>>(bufC, wts, bufD);         // per-mode GEMM
  k_inv_d<<<4608, 128, 0, stream>>>(bufD, tire, tiim, bufB);  // 18432
  k_inv_h<<<13824, 128, 0, stream>>>(bufB, tire, tiim, bufA); // 55296
  k_inv_w<<<41472, 128, 0, stream>>>(bufA, tire, tiim, out);  // 165888
}